// SchemeB_78769700208698
// MI455X (gfx1250) — compile-verified
//
#include <hip/hip_runtime.h>
#include <hip/hip_bf16.h>

#define BB 512
#define GG 978
#define KK 256
#define HH 64
#define NSTEP 6
#define ALPHA 0.1f

typedef float v2f __attribute__((ext_vector_type(2)));
typedef float v8f __attribute__((ext_vector_type(8)));

static __device__ __forceinline__ v8f zero8() {
    v8f z = {0.f, 0.f, 0.f, 0.f, 0.f, 0.f, 0.f, 0.f};
    return z;
}

static __device__ __forceinline__ v8f wmma_f32_4(v2f a, v2f b, v8f c) {
    return __builtin_amdgcn_wmma_f32_16x16x4_f32(
        /*neg_a=*/false, a, /*neg_b=*/false, b,
        /*c_mod=*/(short)0, c, /*reuse_a=*/false, /*reuse_b=*/false);
}

// ---------------------------------------------------------------------------
// Preprocess: rdeg[k] = 1/max(sum_g M[g,k],1);  cew[b] = cell_emb[idx[b]].W_out + b_out
// ---------------------------------------------------------------------------
__global__ __launch_bounds__(256) void prep_kernel(
    const float* __restrict__ M, const float* __restrict__ cell_emb,
    const int* __restrict__ cell_idx, const float* __restrict__ W_out,
    const float* __restrict__ b_out, float* __restrict__ rdeg,
    float* __restrict__ cew) {
    const int t = threadIdx.x;
    float s = 0.f;
    for (int g = 0; g < GG; ++g) s += M[g * KK + t];
    rdeg[t] = 1.0f / fmaxf(s, 1.0f);
    for (int b = t; b < BB; b += 256) {
        const int ci = cell_idx[b];
        float acc = b_out[0];
        for (int h = 0; h < HH; ++h) acc += cell_emb[ci * HH + h] * W_out[h];
        cew[b] = acc;
    }
}

// ---------------------------------------------------------------------------
// Fused per-sample pipeline: one workgroup (8 wave32) per batch sample.
// LDS: one 64KB region reused as (ctl/dt stage) -> (h_mod) -> (Z) -> (zw + W vecs)
// ---------------------------------------------------------------------------
__global__ __launch_bounds__(256) void fused_kernel(
    const float* __restrict__ ctl, const float* __restrict__ dt,
    const float* __restrict__ M, const float* __restrict__ A,
    const float* __restrict__ Wsh, const float* __restrict__ bsh,
    const float* __restrict__ Wmod, const float* __restrict__ bmod,
    const float* __restrict__ Wout, const float* __restrict__ rdeg,
    const float* __restrict__ cew, float* __restrict__ out) {
    __shared__ char smem_raw[65536];
    float* buf = (float*)smem_raw;

    const int b    = blockIdx.x;
    const int tid  = threadIdx.x;
    const int wave = tid >> 5;       // 0..7
    const int lane = tid & 31;
    const int l16  = lane & 15;
    const int hi   = lane >> 4;      // lane-half selects K pair in frags
    const int kt0  = wave * 2;       // this wave owns K-row tiles kt0, kt0+1

    // ---- stage ctl / drug_targets for this sample (pad to multiple of 4) ----
    float* sc = buf;          // [980]
    float* sd = buf + 980;    // [980]
    for (int g = tid; g < 980; g += 256) {
        sc[g] = (g < GG) ? ctl[b * GG + g] : 0.f;
        sd[g] = (g < GG) ? dt[b * GG + g] : 0.f;
    }
    __syncthreads();

    // per-lane weight registers for the 4 owned h-column tiles
    float w0r[4], w1r[4], bsr[4], bmr[4], wor[4];
#pragma unroll
    for (int ht = 0; ht < 4; ++ht) {
        const int h = ht * 16 + l16;
        w0r[ht] = Wsh[h];
        w1r[ht] = Wsh[HH + h];
        bsr[ht] = bsh[h];
        bmr[ht] = bmod[h];
        wor[ht] = Wout[h];
    }

    v8f acc[2][4];
#pragma unroll
    for (int kk = 0; kk < 2; ++kk)
#pragma unroll
        for (int ht = 0; ht < 4; ++ht) acc[kk][ht] = zero8();

    // ---- Phase 1: h_mod = M^T @ relu(x @ W_shared) ----
    // Main body: genes 0..975, no guards -> straight-line loads, no exec branches.
    for (int g0 = 0; g0 < 976; g0 += 4) {
        const int ga = g0 + hi * 2;
        const int gb = ga + 1;
        const float ca = sc[ga], da = sd[ga];
        const float cb = sc[gb], db = sd[gb];
        v2f bf[4];
#pragma unroll
        for (int ht = 0; ht < 4; ++ht) {
            bf[ht].x = fmaxf(ca * w0r[ht] + da * w1r[ht] + bsr[ht], 0.f);
            bf[ht].y = fmaxf(cb * w0r[ht] + db * w1r[ht] + bsr[ht], 0.f);
        }
        v2f af[2];
#pragma unroll
        for (int kk = 0; kk < 2; ++kk) {
            const int k = (kt0 + kk) * 16 + l16;
            af[kk].x = M[ga * KK + k];
            af[kk].y = M[gb * KK + k];
        }
#pragma unroll
        for (int kk = 0; kk < 2; ++kk)
#pragma unroll
            for (int ht = 0; ht < 4; ++ht)
                acc[kk][ht] = wmma_f32_4(af[kk], bf[ht], acc[kk][ht]);
    }
    // Tail: genes 976..979 (978,979 are padding). Clamp addr, cndmask value.
    {
        const int ga = 976 + hi * 2;
        const int gb = ga + 1;
        const float va = (ga < GG) ? 1.f : 0.f;
        const float vb = (gb < GG) ? 1.f : 0.f;
        const int gac = (ga < GG) ? ga : (GG - 1);
        const int gbc = (gb < GG) ? gb : (GG - 1);
        const float ca = sc[ga], da = sd[ga];
        const float cb = sc[gb], db = sd[gb];
        v2f bf[4];
#pragma unroll
        for (int ht = 0; ht < 4; ++ht) {
            bf[ht].x = fmaxf(ca * w0r[ht] + da * w1r[ht] + bsr[ht], 0.f);
            bf[ht].y = fmaxf(cb * w0r[ht] + db * w1r[ht] + bsr[ht], 0.f);
        }
        v2f af[2];
#pragma unroll
        for (int kk = 0; kk < 2; ++kk) {
            const int k = (kt0 + kk) * 16 + l16;
            af[kk].x = va * M[gac * KK + k];
            af[kk].y = vb * M[gbc * KK + k];
        }
#pragma unroll
        for (int kk = 0; kk < 2; ++kk)
#pragma unroll
            for (int ht = 0; ht < 4; ++ht)
                acc[kk][ht] = wmma_f32_4(af[kk], bf[ht], acc[kk][ht]);
    }
    __syncthreads();  // staging region dead; reuse LDS for h_mod [256][64]

#pragma unroll
    for (int kk = 0; kk < 2; ++kk)
#pragma unroll
        for (int ht = 0; ht < 4; ++ht)
#pragma unroll
            for (int r = 0; r < 8; ++r) {
                const int k = (kt0 + kk) * 16 + r + hi * 8;
                const int h = ht * 16 + l16;
                buf[k * HH + h] = acc[kk][ht][r] * rdeg[k];
            }
    __syncthreads();

    // ---- Phase 2: S = relu(h_mod @ W_mod + b_mod) ----
#pragma unroll
    for (int kk = 0; kk < 2; ++kk)
#pragma unroll
        for (int ht = 0; ht < 4; ++ht) acc[kk][ht] = zero8();

    for (int j0 = 0; j0 < HH; j0 += 4) {
        v2f af[2];
#pragma unroll
        for (int kk = 0; kk < 2; ++kk) {
            const int k = (kt0 + kk) * 16 + l16;
            af[kk].x = buf[k * HH + j0 + hi * 2];
            af[kk].y = buf[k * HH + j0 + hi * 2 + 1];
        }
        v2f bf[4];
#pragma unroll
        for (int ht = 0; ht < 4; ++ht) {
            const int h = ht * 16 + l16;
            bf[ht].x = Wmod[(j0 + hi * 2) * HH + h];
            bf[ht].y = Wmod[(j0 + hi * 2 + 1) * HH + h];
        }
#pragma unroll
        for (int kk = 0; kk < 2; ++kk)
#pragma unroll
            for (int ht = 0; ht < 4; ++ht)
                acc[kk][ht] = wmma_f32_4(af[kk], bf[ht], acc[kk][ht]);
    }

    v8f sreg[2][4];  // S stays resident in registers across diffusion
#pragma unroll
    for (int kk = 0; kk < 2; ++kk)
#pragma unroll
        for (int ht = 0; ht < 4; ++ht)
#pragma unroll
            for (int r = 0; r < 8; ++r)
                sreg[kk][ht][r] = fmaxf(acc[kk][ht][r] + bmr[ht], 0.f);

    __syncthreads();  // done reading h_mod; region becomes Z
#pragma unroll
    for (int kk = 0; kk < 2; ++kk)
#pragma unroll
        for (int ht = 0; ht < 4; ++ht)
#pragma unroll
            for (int r = 0; r < 8; ++r) {
                const int k = (kt0 + kk) * 16 + r + hi * 8;
                const int h = ht * 16 + l16;
                buf[k * HH + h] = sreg[kk][ht][r];   // Z := S
            }
    __syncthreads();

    // ---- Phase 3: 6 x  Z = 0.9 * A^T Z + 0.1 * S  (last step stays in regs) ----
    for (int step = 0; step < NSTEP; ++step) {
#pragma unroll
        for (int kk = 0; kk < 2; ++kk)
#pragma unroll
            for (int ht = 0; ht < 4; ++ht) acc[kk][ht] = zero8();

        for (int k0 = 0; k0 < KK; k0 += 4) {
            v2f af[2];
#pragma unroll
            for (int kk = 0; kk < 2; ++kk) {
                const int l = (kt0 + kk) * 16 + l16;     // output row
                af[kk].x = A[(k0 + hi * 2) * KK + l];    // A^T[l,k] = A[k,l]
                af[kk].y = A[(k0 + hi * 2 + 1) * KK + l];
            }
            v2f bf[4];
#pragma unroll
            for (int ht = 0; ht < 4; ++ht) {
                const int h = ht * 16 + l16;
                bf[ht].x = buf[(k0 + hi * 2) * HH + h];
                bf[ht].y = buf[(k0 + hi * 2 + 1) * HH + h];
            }
#pragma unroll
            for (int kk = 0; kk < 2; ++kk)
#pragma unroll
                for (int ht = 0; ht < 4; ++ht)
                    acc[kk][ht] = wmma_f32_4(af[kk], bf[ht], acc[kk][ht]);
        }
        if (step < NSTEP - 1) {
            __syncthreads();
#pragma unroll
            for (int kk = 0; kk < 2; ++kk)
#pragma unroll
                for (int ht = 0; ht < 4; ++ht)
#pragma unroll
                    for (int r = 0; r < 8; ++r) {
                        const int k = (kt0 + kk) * 16 + r + hi * 8;
                        const int h = ht * 16 + l16;
                        buf[k * HH + h] = (1.0f - ALPHA) * acc[kk][ht][r] +
                                          ALPHA * sreg[kk][ht][r];
                    }
            __syncthreads();
        }
    }

    // ---- Phase 4: zw[k] = Z_final[k,:].W_out via in-register shuffle reduce ----
    float zpart[2][8];
#pragma unroll
    for (int kk = 0; kk < 2; ++kk)
#pragma unroll
        for (int r = 0; r < 8; ++r) {
            float p = 0.f;
#pragma unroll
            for (int ht = 0; ht < 4; ++ht) {
                const float z = (1.0f - ALPHA) * acc[kk][ht][r] +
                                ALPHA * sreg[kk][ht][r];
                p += z * wor[ht];   // wor[ht] = W_out[ht*16 + l16]
            }
            zpart[kk][r] = p;
        }
    // butterfly across the 16 lanes sharing the same k (xor 1,2,4,8)
#pragma unroll
    for (int off = 1; off <= 8; off <<= 1)
#pragma unroll
        for (int kk = 0; kk < 2; ++kk)
#pragma unroll
            for (int r = 0; r < 8; ++r)
                zpart[kk][r] += __shfl_xor(zpart[kk][r], off, 32);

    __syncthreads();  // all Z reads from buf complete; alias zw + W vecs
    float* zw  = buf;         // [256]
    float* wv  = buf + 256;   // [64]
    float* w0v = buf + 320;   // [64]
    float* w1v = buf + 384;   // [64]
    float* bsv = buf + 448;   // [64]
    if (l16 == 0) {
#pragma unroll
        for (int kk = 0; kk < 2; ++kk)
#pragma unroll
            for (int r = 0; r < 8; ++r)
                zw[(kt0 + kk) * 16 + r + hi * 8] = zpart[kk][r];
    }
    if (tid < HH) {
        wv[tid]  = Wout[tid];
        w0v[tid] = Wsh[tid];
        w1v[tid] = Wsh[HH + tid];
        bsv[tid] = bsh[tid];
    }
    __syncthreads();

    const float cewb = cew[b];
    for (int g = tid; g < GG; g += 256) {
        const float c = ctl[b * GG + g];
        const float d = dt[b * GG + g];
        float t1 = 0.f;
#pragma unroll
        for (int h = 0; h < HH; ++h)
            t1 += fmaxf(c * w0v[h] + d * w1v[h] + bsv[h], 0.f) * wv[h];
        float hb = 0.f;
        const float4* Mr4 = (const float4*)(M + (size_t)g * KK);
#pragma unroll 4
        for (int k4 = 0; k4 < KK / 4; ++k4) {
            const float4 m = Mr4[k4];
            hb += m.x * zw[k4 * 4 + 0] + m.y * zw[k4 * 4 + 1] +
                  m.z * zw[k4 * 4 + 2] + m.w * zw[k4 * 4 + 3];
        }
        out[b * GG + g] = t1 + hb + cewb;
    }
}

// ---------------------------------------------------------------------------
extern "C" void kernel_launch(void* const* d_in, const int* in_sizes, int n_in,
                              void* d_out, int out_size, void* d_ws, size_t ws_size,
                              hipStream_t stream) {
    const float* ctl      = (const float*)d_in[0];
    const float* drug_tgt = (const float*)d_in[1];
    const int*   cell_idx = (const int*)d_in[2];
    // d_in[3] = drug_fp (unused by the reference)
    const float* M        = (const float*)d_in[4];
    const float* A        = (const float*)d_in[5];
    const float* W_shared = (const float*)d_in[6];
    const float* b_shared = (const float*)d_in[7];
    const float* W_mod    = (const float*)d_in[8];
    const float* b_mod    = (const float*)d_in[9];
    const float* cell_emb = (const float*)d_in[10];
    const float* W_out    = (const float*)d_in[11];
    const float* b_out    = (const float*)d_in[12];

    float* out  = (float*)d_out;
    float* rdeg = (float*)d_ws;          // [256]
    float* cew  = rdeg + KK;             // [512]

    prep_kernel<<<1, 256, 0, stream>>>(M, cell_emb, cell_idx, W_out, b_out,
                                       rdeg, cew);
    fused_kernel<<<BB, 256, 0, stream>>>(ctl, drug_tgt, M, A, W_shared,
                                         b_shared, W_mod, b_mod, W_out, rdeg,
                                         cew, out);
}